// FusedMoE_24275155157411
// MI455X (gfx1250) — compile-verified
//
#include <hip/hip_runtime.h>
#include <cmath>

// ---------------------------------------------------------------------------
// Fused MoE for MI455X (gfx1250, wave32).
// HBM-bound on fp32 weight traffic (805 MB -> ~35 us floor @ 23.3 TB/s), so
// keep fp32 math and use V_WMMA_F32_16X16X4_F32. Experts are looped INSIDE
// each block so one expert's 100.7 MB fp32 weight set stays resident in the
// 192 MB global L2 while all 512 token-tile blocks sweep it. x / out are
// non-temporal so they don't evict the weight stream from L2.
// ---------------------------------------------------------------------------

typedef float v2f __attribute__((ext_vector_type(2)));
typedef float v4f __attribute__((ext_vector_type(4)));   // clang vector: OK for
typedef float v8f __attribute__((ext_vector_type(8)));   // nontemporal builtins

#define T_TOK  8192
#define HDIM   2048
#define IDIM   4096
#define NEXP   8
#define MTILE  16        // tokens per block
#define ICHUNK 128       // intermediate-dim chunk (gate/up cols per pass)
#define XPAD   4         // LDS row padding (floats) -> conflict-free b64 reads
#define APAD   4

// B-operand fetch for V_WMMA_F32_16X16X4_F32.
// B is 4x16 (KxN): VGPR0 lanes0-15 = K0 row, lanes16-31 = K1 row;
//                  VGPR1 lanes0-15 = K2 row, lanes16-31 = K3 row.
// Lane (hi = lane>=16) needs elements k0+hi and k0+2+hi. Use a lane-dependent
// *address* (p+hi) and static extracts [0],[2] of a 12-byte run -> no
// v_cndmask data selection in the inner loop.
struct f3 { float a, b, c; };
__device__ __forceinline__ v2f ld_b(const float* __restrict__ p, int hi) {
  const f3 q = *reinterpret_cast<const f3*>(p + hi);
  return v2f{q.a, q.c};
}

__global__ __launch_bounds__(256) void
moe_fused_f32_wmma(const float* __restrict__ x,
                   const float* __restrict__ logits,
                   const float* __restrict__ w13,
                   const float* __restrict__ w2,
                   float* __restrict__ out)
{
  // 16 x 2048 fp32 token tile (128 KB) + act chunk + routing weights.
  // CDNA5 WGP has 320 KB LDS -> this fits.
  __shared__ float xs[MTILE][HDIM + XPAD];
  __shared__ float act[MTILE][ICHUNK + APAD];
  __shared__ float cw[MTILE][NEXP];

  const int tid  = threadIdx.x;
  const int wave = tid >> 5;          // 8 waves of 32 (wave32)
  const int lane = tid & 31;
  const int lo   = lane & 15;         // M (or N) index within 16
  const int hi   = lane >> 4;         // K-pair select (lanes 16-31 hold K2/K3)
  const int t0   = blockIdx.x * MTILE;

  // ---- stage x tile, coalesced 16B, non-temporal (no reuse after LDS) ----
  for (int idx = tid; idx < MTILE * (HDIM / 4); idx += 256) {
    const int m  = idx / (HDIM / 4);
    const int k4 = idx % (HDIM / 4);
    const v4f q = __builtin_nontemporal_load(
        reinterpret_cast<const v4f*>(x + (size_t)(t0 + m) * HDIM) + k4);
    *reinterpret_cast<v4f*>(&xs[m][k4 * 4]) = q;
  }

  // ---- routing: softmax over 8 logits, top-2, renormalize ----
  if (tid < MTILE) {
    float l[NEXP];
    float mx = -1e30f;
    for (int e = 0; e < NEXP; ++e) {
      l[e] = logits[(size_t)(t0 + tid) * NEXP + e];
      mx = fmaxf(mx, l[e]);
    }
    for (int e = 0; e < NEXP; ++e) l[e] = __expf(l[e] - mx);  // softmax numerators
    int i1 = 0;
    for (int e = 1; e < NEXP; ++e) if (l[e] > l[i1]) i1 = e;
    int i2 = (i1 == 0) ? 1 : 0;
    for (int e = 0; e < NEXP; ++e) if (e != i1 && l[e] > l[i2]) i2 = e;
    const float inv = __builtin_amdgcn_rcpf(l[i1] + l[i2]);
    for (int e = 0; e < NEXP; ++e)
      cw[tid][e] = (e == i1) ? l[i1] * inv : ((e == i2) ? l[i2] * inv : 0.f);
  }
  __syncthreads();

  // Output accumulators: wave owns 256 output columns = 16 WMMA C tiles.
  v8f acc[16];
  #pragma unroll
  for (int j = 0; j < 16; ++j) acc[j] = v8f{0, 0, 0, 0, 0, 0, 0, 0};

  for (int e = 0; e < NEXP; ++e) {
    // Hoist this lane's routing weights (rows i+8*hi) for expert e.
    float cwreg[8];
    #pragma unroll
    for (int i = 0; i < 8; ++i) cwreg[i] = cw[i + 8 * hi][e];

    for (int c = 0; c < IDIM / ICHUNK; ++c) {
      // ======== GEMM1: gate/up 16x16 tiles for this wave, K = HDIM ========
      const size_t grow = (size_t)e * (2 * IDIM) + (size_t)c * ICHUNK
                        + 16 * wave + lo;                 // gate row (w13)
      const float* pg = w13 + grow * HDIM;                // gate row base
      const float* pu = w13 + (grow + (size_t)IDIM) * HDIM;  // up row base

      if (c + 1 < IDIM / ICHUNK) {   // pull next chunk's rows toward L2/L0
        __builtin_prefetch(pg + (size_t)ICHUNK * HDIM, 0, 0);
        __builtin_prefetch(pu + (size_t)ICHUNK * HDIM, 0, 0);
      }

      v8f cg = v8f{0, 0, 0, 0, 0, 0, 0, 0};
      v8f cu = v8f{0, 0, 0, 0, 0, 0, 0, 0};
      #pragma unroll 4
      for (int kk = 0; kk < HDIM; kk += 4) {
        // A (16x4 of x): lane holds m = lo, k = kk + 2*hi + {0,1}
        const v2f a = *reinterpret_cast<const v2f*>(&xs[lo][kk + 2 * hi]);
        const v2f bg = ld_b(pg + kk, hi);
        const v2f bu = ld_b(pu + kk, hi);
        cg = __builtin_amdgcn_wmma_f32_16x16x4_f32(false, a, false, bg,
                                                   (short)0, cg, false, false);
        cu = __builtin_amdgcn_wmma_f32_16x16x4_f32(false, a, false, bu,
                                                   (short)0, cu, false, false);
      }

      // ======== SiLU(gate)*up, scaled by routing weight, into LDS ========
      __syncthreads();   // previous chunk's GEMM2 finished reading act
      #pragma unroll
      for (int i = 0; i < 8; ++i) {
        const int m = i + 8 * hi;            // C tile: rows 0-7 / 8-15 by hi
        const float g = cg[i];
        const float u = cu[i];
        const float s = g * __builtin_amdgcn_rcpf(1.f + __expf(-g));  // SiLU
        act[m][16 * wave + lo] = s * u * cwreg[i];        // fold row scale in
      }
      __syncthreads();   // act chunk ready

      // ======== GEMM2: out(16x2048) += act(16x128) @ w2[e][:,chunk]^T ====
      const size_t w2base = ((size_t)e * HDIM + 256 * wave + lo) * IDIM
                          + (size_t)c * ICHUNK;
      #pragma unroll 2
      for (int kk = 0; kk < ICHUNK; kk += 4) {
        const v2f a = *reinterpret_cast<const v2f*>(&act[lo][kk + 2 * hi]);
        #pragma unroll
        for (int j = 0; j < 16; ++j) {
          const float* pb = w2 + w2base + (size_t)(16 * j) * IDIM + kk;
          const v2f b = ld_b(pb, hi);
          acc[j] = __builtin_amdgcn_wmma_f32_16x16x4_f32(
              false, a, false, b, (short)0, acc[j], false, false);
        }
      }
    }
  }

  // ---- write out: C/D layout -> lane lo = column, VGPR i = row (hi: +8).
  // Non-temporal stores: output is never re-read; keep L2 for weights.
  #pragma unroll
  for (int j = 0; j < 16; ++j) {
    const int n = 256 * wave + 16 * j + lo;
    #pragma unroll
    for (int i = 0; i < 8; ++i) {
      const int m = i + 8 * hi;
      __builtin_nontemporal_store(acc[j][i],
                                  &out[(size_t)(t0 + m) * HDIM + n]);
    }
  }
}

extern "C" void kernel_launch(void* const* d_in, const int* in_sizes, int n_in,
                              void* d_out, int out_size, void* d_ws,
                              size_t ws_size, hipStream_t stream) {
  (void)in_sizes; (void)n_in; (void)out_size; (void)d_ws; (void)ws_size;
  const float* x      = (const float*)d_in[0];
  const float* logits = (const float*)d_in[1];
  const float* w13    = (const float*)d_in[2];
  const float* w2     = (const float*)d_in[3];
  float* out          = (float*)d_out;

  moe_fused_f32_wmma<<<dim3(T_TOK / MTILE), dim3(256), 0, stream>>>(
      x, logits, w13, w2, out);
}